// SliceExctractor_24610162606546
// MI455X (gfx1250) — compile-verified
//
#include <hip/hip_runtime.h>

// Trilinear volume sampling with periodic wrap (mod L-1), 2 channels per point.
// volume: [B=4, 2, 256, 256, 256] f32, grid: [B, 256, 256, 3] f32
// out:    [B, 2, 256, 256] f32
//
// HBM-bound random gather: 16 independent b32 gathers per thread issued as one
// clause for max memory-level parallelism; grid reads (merged to b96) and
// output stores are fully coalesced. Wrap handled with branchless selects.

namespace {
constexpr int      kL    = 256;
constexpr int      kB    = 4;
constexpr int      kWrap = kL - 1;                     // 255
constexpr unsigned kL2   = (unsigned)kL * kL;          // 65536
constexpr unsigned kL3   = (unsigned)kL * kL * kL;     // 16777216
constexpr unsigned kNPts = (unsigned)kB * kL2;         // 262144
}

__global__ __launch_bounds__(256) void trilerp_kernel(
    const float* __restrict__ volume,
    const float* __restrict__ grid,
    float* __restrict__ out)
{
    // Launch is exactly kNPts threads: no bounds check needed.
    const unsigned tid = blockIdx.x * 256u + threadIdx.x;   // b*L*L + y*L + x
    const unsigned b  = tid >> 16;
    const unsigned yx = tid & 0xFFFFu;

    // Coalesced grid read: 12 consecutive bytes per thread (merged to b96).
    const float* gp = grid + (size_t)tid * 3u;
    const float ix = gp[0];
    const float iy = gp[1];
    const float iz = gp[2];

    const float fx = floorf(ix), fy = floorf(iy), fz = floorf(iz);
    const float dx = ix - fx,    dy = iy - fy,    dz = iz - fz;

    // wrap(v) = int(v) % 255 ; inputs lie in [0, 255) so one conditional
    // subtract implements the mod exactly (branchless cndmask).
    int x0 = (int)fx; x0 = (x0 >= kWrap) ? x0 - kWrap : x0;
    int y0 = (int)fy; y0 = (y0 >= kWrap) ? y0 - kWrap : y0;
    int z0 = (int)fz; z0 = (z0 >= kWrap) ? z0 - kWrap : z0;
    int x1 = x0 + 1;  x1 = (x1 >= kWrap) ? 0 : x1;   // ceil neighbor, wrapped
    int y1 = y0 + 1;  y1 = (y1 >= kWrap) ? 0 : y1;
    int z1 = z0 + 1;  z1 = (z1 >= kWrap) ? 0 : z1;

    // Element offsets inside one channel of one batch: y<<16 | x<<8 (+ z).
    const unsigned oy0 = (unsigned)y0 << 16, oy1 = (unsigned)y1 << 16;
    const unsigned ox0 = (unsigned)x0 << 8,  ox1 = (unsigned)x1 << 8;
    const unsigned i00 = oy0 + ox0;   // (y0,x0)
    const unsigned i01 = oy0 + ox1;   // (y0,x1)
    const unsigned i10 = oy1 + ox0;   // (y1,x0)
    const unsigned i11 = oy1 + ox1;   // (y1,x1)

    const float* __restrict__ v0 = volume + (size_t)b * (2u * kL3);  // channel 0
    const float* __restrict__ v1 = v0 + kL3;                          // channel 1

    const unsigned uz0 = (unsigned)z0;
    const unsigned uz1 = (unsigned)z1;

    // Issue all 16 gathers up front (single load clause -> max MLP).
    const float a00l = v0[i00 + uz0];  const float a00h = v0[i00 + uz1];
    const float a01l = v0[i01 + uz0];  const float a01h = v0[i01 + uz1];
    const float a10l = v0[i10 + uz0];  const float a10h = v0[i10 + uz1];
    const float a11l = v0[i11 + uz0];  const float a11h = v0[i11 + uz1];
    const float b00l = v1[i00 + uz0];  const float b00h = v1[i00 + uz1];
    const float b01l = v1[i01 + uz0];  const float b01h = v1[i01 + uz1];
    const float b10l = v1[i10 + uz0];  const float b10h = v1[i10 + uz1];
    const float b11l = v1[i11 + uz0];  const float b11h = v1[i11 + uz1];

    const float wx0 = 1.0f - dx;
    const float wy0 = 1.0f - dy;
    const float wz0 = 1.0f - dz;

    // Channel 0: lerp x, then y, then z (matches reference association).
    {
        const float c00 = a00l * wx0 + a01l * dx;   // (y0, z0)
        const float c10 = a10l * wx0 + a11l * dx;   // (y1, z0)
        const float c0  = c00 * wy0 + c10 * dy;
        const float c01 = a00h * wx0 + a01h * dx;   // (y0, z1)
        const float c11 = a10h * wx0 + a11h * dx;   // (y1, z1)
        const float c1  = c01 * wy0 + c11 * dy;
        out[(size_t)b * (2u * kL2) + yx] = c0 * wz0 + c1 * dz;
    }
    // Channel 1.
    {
        const float c00 = b00l * wx0 + b01l * dx;
        const float c10 = b10l * wx0 + b11l * dx;
        const float c0  = c00 * wy0 + c10 * dy;
        const float c01 = b00h * wx0 + b01h * dx;
        const float c11 = b10h * wx0 + b11h * dx;
        const float c1  = c01 * wy0 + c11 * dy;
        out[(size_t)b * (2u * kL2) + kL2 + yx] = c0 * wz0 + c1 * dz;
    }
}

extern "C" void kernel_launch(void* const* d_in, const int* in_sizes, int n_in,
                              void* d_out, int out_size, void* d_ws, size_t ws_size,
                              hipStream_t stream) {
    (void)in_sizes; (void)n_in; (void)out_size; (void)d_ws; (void)ws_size;
    const float* volume = (const float*)d_in[0];   // [4,2,256,256,256] f32
    const float* grid   = (const float*)d_in[1];   // [4,256,256,3]     f32
    float*       out    = (float*)d_out;           // [4,2,256,256]     f32

    const unsigned nblocks = kNPts / 256u;         // 1024, exact
    trilerp_kernel<<<dim3(nblocks), dim3(256), 0, stream>>>(volume, grid, out);
}